// MLP_view_10007273800070
// MI455X (gfx1250) — compile-verified
//
#include <hip/hip_runtime.h>
#include <math.h>

typedef float v2f __attribute__((ext_vector_type(2)));
typedef float v8f __attribute__((ext_vector_type(8)));

#define DIM 128

// out = relu(X @ W^T + b)
//   X: [N, 128] row-major, W: [128, 128] row-major (torch Linear weight),
//   out: [N, 128].
// Block = 256 threads = 8 waves. Each block computes a 16x128 output tile;
// wave w owns the 16x16 tile at columns [16w, 16w+16).
// WMMA f32 16x16x4 fragment layouts (CDNA5 ISA 7.12.2):
//   A (16x4):  lanes 0-15 hold M=lane, K={k,k+1}; lanes 16-31 hold K={k+2,k+3}
//   B (4x16):  lanes 0-15 hold N=lane, K={k,k+1}; lanes 16-31 hold K={k+2,k+3}
//   C/D 16x16: VGPR r -> M = r (+8 for lanes 16-31), N = lane&15
// Since out[i][j] = dot(X[i,:], W[j,:]), BOTH fragments load row-major rows
// with identical addressing — no transpose needed.
__global__ void __launch_bounds__(256) linear_relu_wmma(
    const float* __restrict__ X, const float* __restrict__ W,
    const float* __restrict__ bias, float* __restrict__ out, int N)
{
    const int lane = threadIdx.x & 31;
    const int wave = threadIdx.x >> 5;       // 0..7 -> column tile
    const int i0   = blockIdx.x * 16;        // row tile base
    const int j0   = wave * 16;              // col tile base

    const int half = lane >> 4;              // 0: lanes 0-15, 1: lanes 16-31
    const int mn   = lane & 15;              // A-row / B-col within tile

    // clamp (keeps loads in-bounds for a ragged last tile; stores are guarded)
    const int arow = (i0 + mn < N) ? (i0 + mn) : (N - 1);

    const float* aptr = X + (size_t)arow * DIM + 2 * half;
    const float* bptr = W + (size_t)(j0 + mn) * DIM + 2 * half;

    v8f c = {0.f, 0.f, 0.f, 0.f, 0.f, 0.f, 0.f, 0.f};

#pragma unroll
    for (int k = 0; k < DIM; k += 4) {
        v2f a = *(const v2f*)(aptr + k);
        v2f b = *(const v2f*)(bptr + k);
        // 8 args: (neg_a, A, neg_b, B, c_mod, C, reuse_a, reuse_b)
        c = __builtin_amdgcn_wmma_f32_16x16x4_f32(
                false, a, false, b, (short)0, c, false, false);
    }

    const float bv = bias[j0 + mn];
#pragma unroll
    for (int r = 0; r < 8; ++r) {
        const int m   = half * 8 + r;        // output row within tile
        const int row = i0 + m;
        if (row < N) {
            float v = c[r] + bv;
            out[(size_t)row * DIM + j0 + mn] = v > 0.f ? v : 0.f;
        }
    }
}

// One edge per wave32. Lane L holds float4 = 4 dims of the 128-dim rows
// (32 lanes x 16B = 512B per row, fully coalesced). tu+tv working set is
// 102 MB -> resident in the 192 MB L2, so the 614 MB of gathers run at L2 BW.
__global__ void __launch_bounds__(256) edge_distance_kernel(
    const float* __restrict__ tu, const float* __restrict__ tv,
    const int* __restrict__ edge_index, const float* __restrict__ edge_val,
    float* __restrict__ out, int E)
{
    const int lane = threadIdx.x & 31;
    const int e = blockIdx.x * (blockDim.x >> 5) + (threadIdx.x >> 5);
    if (e >= E) return;

    const int s = edge_index[e];        // src = edge_index[0][e]
    const int d = edge_index[E + e];    // dst = edge_index[1][e]

    const float4 a = *(const float4*)(tu + (size_t)s * DIM + lane * 4);
    const float4 b = *(const float4*)(tv + (size_t)d * DIM + lane * 4);

    const float dx = a.x - b.x, dy = a.y - b.y;
    const float dz = a.z - b.z, dw = a.w - b.w;
    float sum = dx * dx + dy * dy + dz * dz + dw * dw;

#pragma unroll
    for (int off = 16; off > 0; off >>= 1)
        sum += __shfl_xor(sum, off, 32);

    if (lane == 0) {
        const float dist = sqrtf(sum);
        const float sim  = expf(dist);
        const float sig  = 1.0f / (1.0f + expf(-sim));
        out[e] = edge_val[e] * sig;
    }
}

extern "C" void kernel_launch(void* const* d_in, const int* in_sizes, int n_in,
                              void* d_out, int out_size, void* d_ws, size_t ws_size,
                              hipStream_t stream) {
    const float* Eu = (const float*)d_in[0];
    const float* Ev = (const float*)d_in[1];
    const float* W1 = (const float*)d_in[2];
    const float* b1 = (const float*)d_in[3];
    const float* W2 = (const float*)d_in[4];
    const float* b2 = (const float*)d_in[5];
    const int*   ei = (const int*)d_in[6];   // edge_index, [2, E] flat
    const float* ev = (const float*)d_in[7]; // edge_val, [E]

    float* out = (float*)d_out;

    const int NU = in_sizes[0] / DIM;
    const int NV = in_sizes[1] / DIM;
    const int E  = in_sizes[7];

    // workspace: tu [NU,128] f32 then tv [NV,128] f32 (needs ~102.4 MB)
    float* tu = (float*)d_ws;
    float* tv = tu + (size_t)NU * DIM;

    const int rtilesU = (NU + 15) / 16;
    const int rtilesV = (NV + 15) / 16;
    linear_relu_wmma<<<rtilesU, 256, 0, stream>>>(Eu, W1, b1, tu, NU);
    linear_relu_wmma<<<rtilesV, 256, 0, stream>>>(Ev, W2, b2, tv, NV);

    const int edgesPerBlock = 256 / 32;      // 8 waves -> 8 edges per block
    const int eblocks = (E + edgesPerBlock - 1) / edgesPerBlock;
    edge_distance_kernel<<<eblocks, 256, 0, stream>>>(tu, tv, ei, ev, out, E);
}